// KVCache_18270790877290
// MI455X (gfx1250) — compile-verified
//
#include <hip/hip_runtime.h>
#include <stdint.h>

// KV-cache append: out = [k_cache ++ k_val (seq), v_cache ++ v_val (seq)], fp32.
// B=4 H=32 S=4096 T=1 D=128. Pure memcpy -> HBM-bound (~1.05 GB @ 23.3 TB/s ~ 45us).

typedef float v4f __attribute__((ext_vector_type(4)));

// --- compile-time geometry (vec4 = 16B units) ---
constexpr uint32_t ROW4     = 128 / 4;          // 32 vec4 per D-row
constexpr uint32_t CH_IN4   = 4096u * ROW4;     // 131072 = 2^17 vec4 per (b,h) cache chunk
constexpr uint32_t CH_OUT4  = 4097u * ROW4;     // 131104 vec4 per (b,h) output chunk
constexpr uint32_t BH       = 128;              // B*H
constexpr uint32_t TEN_IN4  = BH * CH_IN4;      // 2^24 vec4 per cache tensor
constexpr uint32_t OUT_T4   = BH * CH_OUT4;     // 16,781,312 vec4 per output tensor

constexpr int      BLOCK     = 256;
constexpr int      GRID_BULK = 8192;
constexpr uint32_t STRIDE    = (uint32_t)BLOCK * (uint32_t)GRID_BULK; // 2^21 threads
constexpr int      ITERS     = (2u * TEN_IN4) / STRIDE;               // 16, exact

// ---------------------------------------------------------------------------
// Bulk: copy both caches into their (strided) positions in the output.
// 16 independent b128 NT loads -> 16 NT stores per thread (deep MLP).
// ---------------------------------------------------------------------------
__global__ __launch_bounds__(BLOCK) void kv_bulk_copy(
    const v4f* __restrict__ kc, const v4f* __restrict__ vc,
    v4f* __restrict__ out) {
  const uint32_t tid0 = blockIdx.x * (uint32_t)BLOCK + threadIdx.x; // < 2^21

  v4f vals[ITERS];
  uint32_t dsts[ITERS];

#pragma unroll
  for (int it = 0; it < ITERS; ++it) {
    // i = tid0 + it*2^21; tensor boundary at 2^24 => tensor = it/8 (compile-time)
    const uint32_t tensor = (uint32_t)(it >> 3);
    const uint32_t j  = tid0 + (uint32_t)it * STRIDE - tensor * TEN_IN4; // < 2^24
    const uint32_t bh = j >> 17;
    const uint32_t r  = j & (CH_IN4 - 1u);
    const v4f* __restrict__ src = (it < 8 ? kc : vc) + j;
    vals[it] = __builtin_nontemporal_load(src);
    dsts[it] = tensor * OUT_T4 + bh * CH_OUT4 + r;
  }
#pragma unroll
  for (int it = 0; it < ITERS; ++it) {
    __builtin_nontemporal_store(vals[it], out + dsts[it]);
  }
}

// ---------------------------------------------------------------------------
// Tail: append the T=1 new K/V rows (2 * 128 * 128 floats = 128 KB) using the
// CDNA5 async LDS DMA path: global_load_async_to_lds_b128 / s_wait_asynccnt /
// global_store_async_from_lds_b128 (ASYNCcnt-tracked, no VGPR data movement).
// 32 blocks * 256 threads = 8192 vec4 = exactly the appended rows.
// ---------------------------------------------------------------------------
__global__ __launch_bounds__(BLOCK) void kv_tail_async(
    const float* __restrict__ k_val, const float* __restrict__ v_val,
    float* __restrict__ out) {
  __shared__ __align__(16) unsigned char smem[BLOCK * 16];

  const uint32_t tid    = threadIdx.x;
  const uint32_t blk    = blockIdx.x;       // 0..31
  const uint32_t tensor = blk >> 4;         // uniform per block
  const uint32_t j      = (blk & 15u) * (uint32_t)BLOCK + tid;  // 0..4095
  const uint32_t bh     = j >> 5;
  const uint32_t c      = j & 31u;

  const float* base = tensor ? v_val : k_val;
  const uint32_t src_off = j * 16u;                                        // bytes
  const uint32_t dst_off = (tensor * OUT_T4 + bh * CH_OUT4 + CH_IN4 + c) * 16u; // bytes, < 2^30

  const uint32_t lds_addr = (uint32_t)(uintptr_t)(&smem[tid * 16]); // low 32b = LDS offset
  const unsigned long long sbase = (unsigned long long)(uintptr_t)base;
  const unsigned long long obase = (unsigned long long)(uintptr_t)out;

  // Memory -> LDS (async, ASYNCcnt++)
  asm volatile("global_load_async_to_lds_b128 %0, %1, %2"
               :: "v"(lds_addr), "v"(src_off), "s"(sbase) : "memory");
  // Wait for LDS to hold the data before the dependent store reads it.
  asm volatile("s_wait_asynccnt 0" ::: "memory");
  // LDS -> Memory (async); S_ENDPGM performs an implicit wait-idle.
  asm volatile("global_store_async_from_lds_b128 %0, %1, %2"
               :: "v"(dst_off), "v"(lds_addr), "s"(obase) : "memory");
  asm volatile("s_wait_asynccnt 0" ::: "memory");
}

// ---------------------------------------------------------------------------
extern "C" void kernel_launch(void* const* d_in, const int* in_sizes, int n_in,
                              void* d_out, int out_size, void* d_ws, size_t ws_size,
                              hipStream_t stream) {
  const v4f*   kc = (const v4f*)d_in[0];
  const v4f*   vc = (const v4f*)d_in[1];
  const float* kv = (const float*)d_in[2];
  const float* vv = (const float*)d_in[3];

  kv_bulk_copy<<<GRID_BULK, BLOCK, 0, stream>>>(kc, vc, (v4f*)d_out);
  kv_tail_async<<<32, BLOCK, 0, stream>>>(kv, vv, (float*)d_out);
}